// Model_PDE_3_67362267070603
// MI455X (gfx1250) — compile-verified
//
#include <hip/hip_runtime.h>
#include <math.h>

typedef __attribute__((ext_vector_type(2))) float v2f;
typedef __attribute__((ext_vector_type(8))) float v8f;

#define HIDDEN 20
#define NMID 8
#define HP 32              // hidden padded to two 16-row WMMA M-tiles
#define TILE_N 16          // points per wave
#define WAVES_PER_BLOCK 4
#define THREADS (WAVES_PER_BLOCK * 32)

// PDE constants from the reference
#define C_B   0.5f
#define C_C   0.5f
#define C_SIG 1.0f
#define C_BF  0.5f
#define C_CF  0.9f

// Hardware tanh: V_TANH_F32 (TRANS32). Builtin if available, else inline asm
// with a trailing v_nop to satisfy the TRANS multicycle RAW hazard rule.
__device__ __forceinline__ float fast_tanh(float v) {
#if __has_builtin(__builtin_amdgcn_tanhf)
    return __builtin_amdgcn_tanhf(v);
#elif __has_builtin(__builtin_amdgcn_tanh_f32)
    return __builtin_amdgcn_tanh_f32(v);
#else
    float r;
    asm volatile("v_tanh_f32 %0, %1\n\tv_nop" : "=v"(r) : "v"(v));
    return r;
#endif
}

__global__ __launch_bounds__(THREADS) void pinn_fwd_taylor_kernel(
    const float* __restrict__ x,     // [n,2]
    const float* __restrict__ Win,   // [20,2]
    const float* __restrict__ bin,   // [20]
    const float* __restrict__ Wmid,  // [8,20,20]
    const float* __restrict__ bmid,  // [8,20]
    const float* __restrict__ Wout,  // [1,20]
    const float* __restrict__ bout,  // [1]
    float* __restrict__ out,         // [2*n]: f then pde
    int n)
{
    // ---- LDS: padded weights/biases + per-wave Taylor state + x tile ----
    __shared__ float sW[NMID][HP][HIDDEN];                 // rows 20..31 zero
    __shared__ float sB[NMID][HP];                         // rows 20..31 zero
    __shared__ float sWin[HIDDEN][2];
    __shared__ float sbin[HIDDEN];
    __shared__ float sWout[HIDDEN];
    __shared__ float sbout;
    __shared__ float sS[WAVES_PER_BLOCK][4][HP][TILE_N];   // [tangent][k][point]
    __shared__ float sX[WAVES_PER_BLOCK][2 * TILE_N];      // interleaved x0,x1

    const int tid = threadIdx.x;

    for (int i = tid; i < NMID * HP * HIDDEN; i += THREADS) {
        int l = i / (HP * HIDDEN);
        int r = (i / HIDDEN) % HP;
        int c = i % HIDDEN;
        sW[l][r][c] = (r < HIDDEN) ? Wmid[(l * HIDDEN + r) * HIDDEN + c] : 0.0f;
    }
    for (int i = tid; i < NMID * HP; i += THREADS) {
        int l = i / HP, r = i % HP;
        sB[l][r] = (r < HIDDEN) ? bmid[l * HIDDEN + r] : 0.0f;
    }
    for (int i = tid; i < HIDDEN; i += THREADS) {
        sWin[i][0] = Win[2 * i + 0];
        sWin[i][1] = Win[2 * i + 1];
        sbin[i]    = bin[i];
        sWout[i]   = Wout[i];
    }
    if (tid == 0) sbout = bout[0];
    __syncthreads();

    const int wave  = tid >> 5;
    const int lane  = tid & 31;
    const int ln16  = lane & 15;
    const int hi    = lane >> 4;                 // 0: lanes 0-15, 1: lanes 16-31
    const int pbase = (blockIdx.x * WAVES_PER_BLOCK + wave) * TILE_N;
    if (pbase >= n) return;                      // wave-uniform, after barrier

    // stage this wave's 16 points (one coalesced float per lane)
    sX[wave][lane] = x[2 * pbase + lane];

    float (*S)[HP][TILE_N] = sS[wave];

    // ---- input layer: z = Win @ xi + bin; seed tangents (t, x, xx) ----
    for (int i = lane; i < HP * TILE_N; i += 32) {   // 512/32 = 16 iters exact
        int k = i / TILE_N, nn = i % TILE_N;
        float v0 = 0.f, vt = 0.f, vx = 0.f, vxx = 0.f;
        if (k < HIDDEN) {
            float x0 = sX[wave][2 * nn + 0];
            float x1 = sX[wave][2 * nn + 1];
            float w0 = sWin[k][0], w1 = sWin[k][1];
            float z  = fmaf(w0, x0, fmaf(w1, x1, sbin[k]));
            float a  = fast_tanh(z);
            float s  = fmaf(-a, a, 1.0f);
            v0 = a; vt = s * w0; vx = s * w1;
            vxx = -2.0f * a * vx * w1;           // z_xx = 0, z_x = w1
        }
        S[0][k][nn] = v0; S[1][k][nn] = vt; S[2][k][nn] = vx; S[3][k][nn] = vxx;
    }

    // ---- 8 mid layers: 32x20 @ 20x16 GEMM per tangent via f32 WMMA ----
    for (int layer = 0; layer < NMID; ++layer) {
        v8f acc[2][4];
        #pragma unroll
        for (int mt = 0; mt < 2; ++mt) {
            #pragma unroll
            for (int r = 0; r < 8; ++r)
                acc[mt][0][r] = sB[layer][mt * 16 + r + hi * 8];  // bias in C
            #pragma unroll
            for (int t = 1; t < 4; ++t)
                #pragma unroll
                for (int r = 0; r < 8; ++r)
                    acc[mt][t][r] = 0.0f;
        }

        #pragma unroll
        for (int kb = 0; kb < 5; ++kb) {                 // K = 20 = 5 x 4
            int k0 = kb * 4 + hi * 2;                    // lanes 16-31: K+2,K+3
            v2f Bop[4];
            #pragma unroll
            for (int t = 0; t < 4; ++t) {
                Bop[t][0] = S[t][k0 + 0][ln16];
                Bop[t][1] = S[t][k0 + 1][ln16];
            }
            v2f Aop[2];
            #pragma unroll
            for (int mt = 0; mt < 2; ++mt) {
                int m = mt * 16 + ln16;
                Aop[mt][0] = sW[layer][m][k0 + 0];
                Aop[mt][1] = sW[layer][m][k0 + 1];
            }
            #pragma unroll
            for (int mt = 0; mt < 2; ++mt)
                #pragma unroll
                for (int t = 0; t < 4; ++t)
                    acc[mt][t] = __builtin_amdgcn_wmma_f32_16x16x4_f32(
                        false, Aop[mt], false, Bop[t],
                        (short)0, acc[mt][t], false, false);
        }

        // tanh chain on D tiles, write state back for next layer
        #pragma unroll
        for (int mt = 0; mt < 2; ++mt) {
            #pragma unroll
            for (int r = 0; r < 8; ++r) {
                int m = mt * 16 + r + hi * 8;
                float zv  = acc[mt][0][r], zt = acc[mt][1][r];
                float zx  = acc[mt][2][r], zxx = acc[mt][3][r];
                float a   = fast_tanh(zv);
                float s   = fmaf(-a, a, 1.0f);
                float at  = s * zt;
                float ax  = s * zx;
                float axx = fmaf(s, zxx, -2.0f * a * ax * zx);
                S[0][m][ln16] = a;  S[1][m][ln16] = at;
                S[2][m][ln16] = ax; S[3][m][ln16] = axx;
            }
        }
    }

    // ---- output layer + PDE residual (lanes 0-15, one point each) ----
    if (lane < TILE_N) {
        int p = pbase + lane;
        float f = sbout, ft = 0.f, fx = 0.f, fxx = 0.f;
        #pragma unroll
        for (int k = 0; k < HIDDEN; ++k) {
            float w = sWout[k];
            f   = fmaf(w, S[0][k][lane], f);
            ft  = fmaf(w, S[1][k][lane], ft);
            fx  = fmaf(w, S[2][k][lane], fx);
            fxx = fmaf(w, S[3][k][lane], fxx);
        }
        float xs  = sX[wave][2 * lane + 1];
        float pde = C_BF * xs * xs + ft + 0.5f * C_SIG * C_SIG * fxx
                  + C_B * xs * fx
                  - (C_C * C_C) / (4.0f * C_CF) * fx * fx;
        out[p]     = f;
        out[n + p] = pde;
    }
}

extern "C" void kernel_launch(void* const* d_in, const int* in_sizes, int n_in,
                              void* d_out, int out_size, void* d_ws, size_t ws_size,
                              hipStream_t stream) {
    const float* x    = (const float*)d_in[0];
    const float* Win  = (const float*)d_in[1];
    const float* bin  = (const float*)d_in[2];
    const float* Wmid = (const float*)d_in[3];
    const float* bmid = (const float*)d_in[4];
    const float* Wout = (const float*)d_in[5];
    const float* bout = (const float*)d_in[6];
    float* out = (float*)d_out;

    int n = in_sizes[0] / 2;                       // x is [n,2]
    int ptsPerBlock = TILE_N * WAVES_PER_BLOCK;    // 64
    int blocks = (n + ptsPerBlock - 1) / ptsPerBlock;
    pinn_fwd_taylor_kernel<<<blocks, THREADS, 0, stream>>>(
        x, Win, bin, Wmid, bmid, Wout, bout, out, n);
}